// GCN_85177791415007
// MI455X (gfx1250) — compile-verified
//
#include <hip/hip_runtime.h>
#include <math.h>

typedef __attribute__((ext_vector_type(2))) float v2f;
typedef __attribute__((ext_vector_type(8))) float v8f;

#define DIM_IN 256
#define DIM_H  128

// ---------------------------------------------------------------- degree ----
__global__ void deg_init_k(float* __restrict__ deg, int n) {
    int i = blockIdx.x * blockDim.x + threadIdx.x;
    if (i < n) deg[i] = 1.0f;  // self-loop contribution
}

__global__ void deg_edges_k(const long long* __restrict__ dst,
                            float* __restrict__ deg, int E) {
    int e = blockIdx.x * blockDim.x + threadIdx.x;
    if (e < E) atomicAdd(&deg[(int)dst[e]], 1.0f);
}

__global__ void deg_rsqrt_k(float* __restrict__ deg, int n) {
    int i = blockIdx.x * blockDim.x + threadIdx.x;
    if (i < n) {
        float d = deg[i];
        deg[i] = (d > 0.0f) ? rsqrtf(d) : 0.0f;  // becomes dinv in-place
    }
}

// ------------------------------------------------- GEMM: H = X @ W1 (WMMA) --
// Grid: N/16 blocks of 256 threads (8 waves). Wave w computes the 16x16 tile
// at rows [16*blockIdx.x, +16), cols [16*w, +16). K=256 via 64 chained
// v_wmma_f32_16x16x4_f32. Layouts per CDNA5 ISA 7.12.2 (32-bit A 16x4,
// 32-bit C/D 16x16).
__global__ void __launch_bounds__(256)
gemm_wmma_f32_k(const float* __restrict__ X, const float* __restrict__ W,
                float* __restrict__ H) {
    const int wave = threadIdx.x >> 5;
    const int lane = threadIdx.x & 31;
    const int lane16 = lane & 15;
    const bool hi = lane >= 16;

    const int m0 = blockIdx.x * 16;
    const int n0 = wave * 16;
    const int col = n0 + lane16;

    const float* __restrict__ xrow = X + (size_t)(m0 + lane16) * DIM_IN;

    v8f acc = {};
#pragma unroll 8
    for (int k = 0; k < DIM_IN; k += 4) {
        const int ka = k + (hi ? 2 : 0);
        // A fragment: lane L<16 holds A[M=L][ka], A[M=L][ka+1]; hi half K+2/K+3
        v2f a;
        a.x = xrow[ka];
        a.y = xrow[ka + 1];
        // B fragment: lane L<16 holds B[K=ka][n0+L], B[K=ka+1][n0+L]
        v2f b;
        b.x = W[(size_t)ka * DIM_H + col];
        b.y = W[(size_t)(ka + 1) * DIM_H + col];
        acc = __builtin_amdgcn_wmma_f32_16x16x4_f32(
            /*neg_a=*/false, a, /*neg_b=*/false, b,
            /*c_mod=*/(short)0, acc, /*reuse_a=*/false, /*reuse_b=*/false);
    }

    // D layout: VGPR r -> M = r (lanes 0-15) or 8+r (lanes 16-31), N = lane&15
    const int mbase = m0 + (hi ? 8 : 0);
#pragma unroll
    for (int r = 0; r < 8; ++r) {
        H[(size_t)(mbase + r) * DIM_H + col] = acc[r];
    }
}

// ------------------------------------------ layer-1 aggregation (128-wide) --
// out1[i][c] = dinv[i]^2 * H[i][c] + b1[c]   (self-loop + bias init)
__global__ void agg1_init_k(const float* __restrict__ H,
                            const float* __restrict__ dinv,
                            const float* __restrict__ b1,
                            float* __restrict__ out1, int n) {
    int idx = blockIdx.x * blockDim.x + threadIdx.x;
    if (idx < n * DIM_H) {
        int node = idx >> 7;
        int c = idx & (DIM_H - 1);
        float di = dinv[node];
        out1[idx] = di * di * H[idx] + b1[c];
    }
}

// One wave per edge: coalesced float4 gather of H[src], 4 f32 atomics/lane
// into out1[dst] (dst matrix is L2-resident: 51MB << 192MB L2).
__global__ void __launch_bounds__(256)
agg1_edges_k(const float* __restrict__ H, const float* __restrict__ dinv,
             const long long* __restrict__ src, const long long* __restrict__ dst,
             float* __restrict__ out1, int E) {
    int e = (int)((blockIdx.x * blockDim.x + threadIdx.x) >> 5);
    if (e >= E) return;
    int lane = threadIdx.x & 31;
    int s = (int)src[e];
    int d = (int)dst[e];
    float norm = dinv[s] * dinv[d];
    const float4* __restrict__ hs = (const float4*)(H + (size_t)s * DIM_H);
    float* __restrict__ od = out1 + (size_t)d * DIM_H + lane * 4;
    float4 v = hs[lane];
    atomicAdd(od + 0, norm * v.x);
    atomicAdd(od + 1, norm * v.y);
    atomicAdd(od + 2, norm * v.z);
    atomicAdd(od + 3, norm * v.w);
}

// --------------------------- fused ReLU + GEMV: z2[i] = relu(out1[i]) @ W2 --
__global__ void __launch_bounds__(256)
gemv_relu_k(const float* __restrict__ out1, const float* __restrict__ W2,
            float* __restrict__ z2, int n) {
    int node = (int)((blockIdx.x * blockDim.x + threadIdx.x) >> 5);
    if (node >= n) return;
    int lane = threadIdx.x & 31;
    const float4* __restrict__ row = (const float4*)(out1 + (size_t)node * DIM_H);
    const float4* __restrict__ w = (const float4*)W2;
    float4 v = row[lane];
    float4 wv = w[lane];
    float acc = fmaxf(v.x, 0.0f) * wv.x + fmaxf(v.y, 0.0f) * wv.y +
                fmaxf(v.z, 0.0f) * wv.z + fmaxf(v.w, 0.0f) * wv.w;
#pragma unroll
    for (int off = 16; off > 0; off >>= 1)
        acc += __shfl_xor(acc, off, 32);
    if (lane == 0) z2[node] = acc;
}

// ----------------------------------------------------- layer-2 aggregation --
__global__ void agg2_init_k(const float* __restrict__ z2,
                            const float* __restrict__ dinv,
                            const float* __restrict__ b2,
                            float* __restrict__ out, int n) {
    int i = blockIdx.x * blockDim.x + threadIdx.x;
    if (i < n) {
        float di = dinv[i];
        out[i] = di * di * z2[i] + b2[0];
    }
}

__global__ void agg2_edges_k(const float* __restrict__ z2,
                             const float* __restrict__ dinv,
                             const long long* __restrict__ src,
                             const long long* __restrict__ dst,
                             float* __restrict__ out, int E) {
    int e = blockIdx.x * blockDim.x + threadIdx.x;
    if (e < E) {
        int s = (int)src[e];
        int d = (int)dst[e];
        atomicAdd(&out[d], dinv[s] * dinv[d] * z2[s]);
    }
}

__global__ void sigmoid_k(float* __restrict__ out, int n) {
    int i = blockIdx.x * blockDim.x + threadIdx.x;
    if (i < n) out[i] = 1.0f / (1.0f + expf(-out[i]));
}

// ---------------------------------------------------------------- launch ----
extern "C" void kernel_launch(void* const* d_in, const int* in_sizes, int n_in,
                              void* d_out, int out_size, void* d_ws, size_t ws_size,
                              hipStream_t stream) {
    const float* x = (const float*)d_in[0];           // [N, 256]
    const long long* eidx = (const long long*)d_in[1];// [2, E] int64
    const float* W1 = (const float*)d_in[2];          // [256, 128]
    const float* b1 = (const float*)d_in[3];          // [128]
    const float* W2 = (const float*)d_in[4];          // [128]
    const float* b2 = (const float*)d_in[5];          // [1]

    const int N = in_sizes[0] / DIM_IN;   // 100000
    const int E = in_sizes[1] / 2;        // 1600000
    const long long* src = eidx;
    const long long* dst = eidx + E;

    // workspace layout (floats): dinv[N] | H1[N*128] | out1[N*128] | z2[N]
    float* dinv = (float*)d_ws;
    float* H1   = dinv + N;
    float* out1 = H1 + (size_t)N * DIM_H;
    float* z2   = out1 + (size_t)N * DIM_H;
    float* out  = (float*)d_out;

    const int T = 256;

    // degrees -> dinv (in place)
    deg_init_k<<<(N + T - 1) / T, T, 0, stream>>>(dinv, N);
    deg_edges_k<<<(E + T - 1) / T, T, 0, stream>>>(dst, dinv, E);
    deg_rsqrt_k<<<(N + T - 1) / T, T, 0, stream>>>(dinv, N);

    // H1 = X @ W1 via v_wmma_f32_16x16x4_f32 (N % 16 == 0)
    gemm_wmma_f32_k<<<N / 16, 256, 0, stream>>>(x, W1, H1);

    // layer-1 aggregation
    agg1_init_k<<<(N * DIM_H + T - 1) / T, T, 0, stream>>>(H1, dinv, b1, out1, N);
    {
        long long waves = (long long)E;                   // one wave per edge
        long long blocks = (waves * 32 + T - 1) / T;
        agg1_edges_k<<<(int)blocks, T, 0, stream>>>(H1, dinv, src, dst, out1, E);
    }

    // z2 = relu(out1) @ W2  (one wave per node)
    {
        long long blocks = ((long long)N * 32 + T - 1) / T;
        gemv_relu_k<<<(int)blocks, T, 0, stream>>>(out1, W2, z2, N);
    }

    // layer-2 aggregation directly into d_out, then sigmoid in place
    agg2_init_k<<<(N + T - 1) / T, T, 0, stream>>>(z2, dinv, b2, out, N);
    agg2_edges_k<<<(E + T - 1) / T, T, 0, stream>>>(z2, dinv, src, dst, out, E);
    sigmoid_k<<<(N + T - 1) / T, T, 0, stream>>>(out, N);
}